// GCN_1675037246076
// MI455X (gfx1250) — compile-verified
//
#include <hip/hip_runtime.h>
#include <stdint.h>

// GCN (2x GCNConv + linear head) for N=100000 nodes, E=3200000 edges.
// Dense GEMMs via V_WMMA_F32_16X16X4_F32 (fp32-exact), sparse aggregation via
// L2-resident fp32 global atomics with float4 gathers. Self-loop + bias folded
// into GEMM epilogue; ReLU folded into next GEMM's A-load.

typedef __attribute__((ext_vector_type(2))) float v2f;
typedef __attribute__((ext_vector_type(8))) float v8f;

#define NNODES 100000
#define NEDGES 3200000

__global__ void zero_f32_kernel(float* __restrict__ p, int n) {
  int i = blockIdx.x * blockDim.x + threadIdx.x;
  if (i < n) p[i] = 0.0f;
}

__global__ void deg_count_kernel(const long long* __restrict__ dst,
                                 float* __restrict__ deg, int e) {
  int i = blockIdx.x * blockDim.x + threadIdx.x;
  if (i < e) atomicAdd(&deg[(int)dst[i]], 1.0f);
}

// dinv = rsqrt(in_degree + 1)   (+1 = self loop; always > 0)
__global__ void make_dinv_kernel(float* __restrict__ degdinv, int n) {
  int i = blockIdx.x * blockDim.x + threadIdx.x;
  if (i < n) degdinv[i] = rsqrtf(degdinv[i] + 1.0f);
}

__global__ void edge_norm_kernel(const long long* __restrict__ src,
                                 const long long* __restrict__ dst,
                                 const float* __restrict__ dinv,
                                 float* __restrict__ nrm, int e) {
  int i = blockIdx.x * blockDim.x + threadIdx.x;
  if (i < e) nrm[i] = dinv[(int)src[i]] * dinv[(int)dst[i]];
}

// H = act(A) @ W  computed with V_WMMA_F32_16X16X4_F32, one wave per 16-row tile.
// FINAL=false: H written raw; AGG = bias + dinv[row]^2 * H  (self-loop + bias init).
// FINAL=true : H = H + bias written directly (model output).
template <int K, int NOUT, bool RELU_IN, bool FINAL>
__global__ void gemm_wmma_kernel(const float* __restrict__ A,
                                 const float* __restrict__ W,
                                 const float* __restrict__ bias,
                                 const float* __restrict__ dinv,
                                 float* __restrict__ H,
                                 float* __restrict__ AGG,
                                 int ntiles) {
  const int wave = threadIdx.x >> 5;
  const int tile = blockIdx.x * (blockDim.x >> 5) + wave;
  if (tile >= ntiles) return;              // wave-uniform: EXEC stays all-ones

  const int lane = threadIdx.x & 31;
  const int half = lane >> 4;              // 0: lanes 0-15, 1: lanes 16-31
  const int r    = lane & 15;
  const int row0 = tile * 16;
  constexpr int NT = NOUT / 16;

  v8f acc[NT] = {};

  const float* __restrict__ Arow = A + (size_t)(row0 + r) * K;

#pragma unroll
  for (int kk = 0; kk < K; kk += 4) {
    // A 16x4 fp32 layout: lanes 0-15 -> K = kk+{0,1}; lanes 16-31 -> K = kk+{2,3}
    float a0 = Arow[kk + 2 * half + 0];
    float a1 = Arow[kk + 2 * half + 1];
    if (RELU_IN) { a0 = fmaxf(a0, 0.0f); a1 = fmaxf(a1, 0.0f); }
    v2f a; a.x = a0; a.y = a1;
#pragma unroll
    for (int t = 0; t < NT; ++t) {
      const int col = t * 16 + r;
      // B 4x16 fp32 layout mirrors A: v0/v1 rows K = kk+2*half+{0,1}
      v2f b;
      b.x = W[(size_t)(kk + 2 * half + 0) * NOUT + col];
      b.y = W[(size_t)(kk + 2 * half + 1) * NOUT + col];
      acc[t] = __builtin_amdgcn_wmma_f32_16x16x4_f32(
          /*neg_a=*/false, a, /*neg_b=*/false, b,
          /*c_mod=*/(short)0, acc[t], /*reuse_a=*/false, /*reuse_b=*/false);
    }
  }

#pragma unroll
  for (int t = 0; t < NT; ++t) {
    const int col = t * 16 + r;
#pragma unroll
    for (int vr = 0; vr < 8; ++vr) {
      const int row = row0 + vr + 8 * half;   // C/D layout: rows 0-7 / 8-15
      const float v = acc[t][vr];
      if (FINAL) {
        H[(size_t)row * NOUT + col] = v + bias[col];
      } else {
        H[(size_t)row * NOUT + col] = v;
        const float di = dinv[row];
        AGG[(size_t)row * NOUT + col] = bias[col] + di * di * v;
      }
    }
  }
}

// AGG[d] += norm[e] * H[s]; one thread per (edge, 4-feature chunk).
// float4 gather (global_load_b128) + 4 fp32 atomics; index/nrm loads amortized 4x.
template <int F>
__global__ void scatter_edges_kernel(const long long* __restrict__ src,
                                     const long long* __restrict__ dst,
                                     const float* __restrict__ nrm,
                                     const float* __restrict__ H,
                                     float* __restrict__ AGG) {
  constexpr int FQ = F / 4;
  const long long idx = (long long)blockIdx.x * blockDim.x + threadIdx.x;
  if (idx >= (long long)NEDGES * FQ) return;
  const int e = (int)(idx / FQ);
  const int q = (int)(idx - (long long)e * FQ);
  const int s = (int)src[e];
  const int d = (int)dst[e];
  const float w = nrm[e];
  const float4 hv = *reinterpret_cast<const float4*>(H + (size_t)s * F + q * 4);
  float* ap = AGG + (size_t)d * F + q * 4;
  atomicAdd(ap + 0, w * hv.x);
  atomicAdd(ap + 1, w * hv.y);
  atomicAdd(ap + 2, w * hv.z);
  atomicAdd(ap + 3, w * hv.w);
}

extern "C" void kernel_launch(void* const* d_in, const int* in_sizes, int n_in,
                              void* d_out, int out_size, void* d_ws, size_t ws_size,
                              hipStream_t stream) {
  const float*     x    = (const float*)d_in[0];          // [N,16]
  const long long* ei   = (const long long*)d_in[1];      // [2,E] int64
  const float*     W1   = (const float*)d_in[2];          // [16,48]
  const float*     b1   = (const float*)d_in[3];          // [48]
  const float*     W2   = (const float*)d_in[4];          // [48,32]
  const float*     b2   = (const float*)d_in[5];          // [32]
  const float*     Wfc  = (const float*)d_in[6];          // [32,16]
  const float*     bfc  = (const float*)d_in[7];          // [16]
  float*           out  = (float*)d_out;                  // [N,16]

  const long long* srcp = ei;            // row 0
  const long long* dstp = ei + NEDGES;   // row 1

  // Workspace layout (floats): 19.3M floats ~= 77.2 MB
  float* ws   = (float*)d_ws;
  float* dinv = ws;                         // N
  float* nrm  = dinv + NNODES;              // E
  float* h1   = nrm  + NEDGES;              // N*48
  float* agg1 = h1   + (size_t)NNODES * 48; // N*48
  float* h2   = agg1 + (size_t)NNODES * 48; // N*32
  float* agg2 = h2   + (size_t)NNODES * 32; // N*32

  const int B = 256;
  const int gN = (NNODES + B - 1) / B;
  const int gE = (NEDGES + B - 1) / B;

  // Graph normalization (computed fresh every call; shared by both layers)
  zero_f32_kernel<<<gN, B, 0, stream>>>(dinv, NNODES);
  deg_count_kernel<<<gE, B, 0, stream>>>(dstp, dinv, NEDGES);
  make_dinv_kernel<<<gN, B, 0, stream>>>(dinv, NNODES);
  edge_norm_kernel<<<gE, B, 0, stream>>>(srcp, dstp, dinv, nrm, NEDGES);

  const int ntiles = NNODES / 16;                 // 6250 exactly
  const int gG = (ntiles + (B / 32) - 1) / (B / 32);

  // Layer 1: h1 = x@W1 ; agg1 = b1 + dinv^2*h1 ; agg1[d] += nrm*h1[s]
  gemm_wmma_kernel<16, 48, false, false><<<gG, B, 0, stream>>>(x, W1, b1, dinv, h1, agg1, ntiles);
  {
    const long long tot = (long long)NEDGES * (48 / 4);
    scatter_edges_kernel<48><<<(int)((tot + B - 1) / B), B, 0, stream>>>(srcp, dstp, nrm, h1, agg1);
  }

  // Layer 2: h2 = relu(agg1)@W2 ; agg2 = b2 + dinv^2*h2 ; agg2[d] += nrm*h2[s]
  gemm_wmma_kernel<48, 32, true, false><<<gG, B, 0, stream>>>(agg1, W2, b2, dinv, h2, agg2, ntiles);
  {
    const long long tot = (long long)NEDGES * (32 / 4);
    scatter_edges_kernel<32><<<(int)((tot + B - 1) / B), B, 0, stream>>>(srcp, dstp, nrm, h2, agg2);
  }

  // Head: out = relu(agg2)@Wfc + bfc
  gemm_wmma_kernel<32, 16, true, true><<<gG, B, 0, stream>>>(agg2, Wfc, bfc, dinv, out, nullptr, ntiles);
}